// Net_16896401343212
// MI455X (gfx1250) — compile-verified
//
#include <hip/hip_runtime.h>

// ---------------------------------------------------------------------------
// Sparse 3x3x3 conv scatter for MI455X (gfx1250, wave32).
//
// Pipeline (memory-bound; ~11 GFLOP of GEMM is noise):
//   1. zero 998x998x38 u32 count grid (151 MB)
//   2. pre-fill output: rows=0, uniq=SENT
//   3. occupancy pass: per (point, offset) atomic count into grid
//   4. block counts -> 5. exclusive block offsets (sorted compaction)
//   6. compact: write sorted uniq, convert count grid to cell->row map
//      in place, seed occupied rows with bias
//   7. WMMA conv: 16x64 f32 tile (async-loaded to LDS) x 27 weight mats
//      (async-loaded to LDS), scatter f32 atomics straight into final rows
// ---------------------------------------------------------------------------

typedef float v2f __attribute__((ext_vector_type(2)));
typedef float v8f __attribute__((ext_vector_type(8)));

#define NPTS   200000
#define CIN    64
#define COUT   16
#define KVOL   27
#define SP0    998
#define SP1    998
#define SP2    38
#define GRIDC  37848152LL          /* 998*998*38 */
#define SENTV  37848152            /* sentinel linear index */
#define NKTOT  5400000LL           /* NPTS*KVOL */
#define TILES  12500               /* NPTS/16 */
#define WPB    8                   /* waves per block */
#define SCAN_CELLS 4096
#define SCAN_THREADS 256
#define NB     9241                /* ceil(GRIDC / SCAN_CELLS) */

// ---- CDNA5 async global->LDS helpers (ASYNCcnt-tracked LDS DMA) -----------

__device__ __forceinline__ unsigned lds_off(const void* p) {
  // generic -> LDS address space, then ptrtoint gives the 32-bit LDS offset
  return (unsigned)(unsigned long long)(__attribute__((address_space(3))) const void*)p;
}

__device__ __forceinline__ void async_ld_b128(unsigned lds, const void* g) {
  asm volatile("global_load_async_to_lds_b128 %0, %1, off"
               :: "v"(lds), "v"((unsigned long long)g) : "memory");
}

__device__ __forceinline__ void wait_async0() {
  asm volatile("s_wait_asynccnt 0x0" ::: "memory");
}

// ---------------- workspace / output initialization ----------------

__global__ void zero_cnt_kernel(uint4* __restrict__ p, long long n4) {
  long long i = (long long)blockIdx.x * blockDim.x + threadIdx.x;
  long long stride = (long long)gridDim.x * blockDim.x;
  uint4 z; z.x = 0u; z.y = 0u; z.z = 0u; z.w = 0u;
  for (; i < n4; i += stride) p[i] = z;
}

__global__ void fill_out_kernel(int* __restrict__ out, long long zero_words,
                                long long total_words, int sent) {
  long long i = (long long)blockIdx.x * blockDim.x + threadIdx.x;
  long long stride = (long long)gridDim.x * blockDim.x;
  for (; i < total_words; i += stride)
    out[i] = (i < zero_words) ? 0 : sent;   // 0.0f bits == 0
}

// ---------------- occupancy pass ----------------

__global__ void count_pass(const int* __restrict__ coords,
                           unsigned* __restrict__ cnt) {
  const int n = blockIdx.x * blockDim.x + threadIdx.x;
  if (n >= NPTS) return;
  const int* cp = coords + (size_t)n * 4;
  __builtin_prefetch(cp, 0, 1);
  const int x = cp[1], y = cp[2], z = cp[3];
#pragma unroll
  for (int k = 0; k < KVOL; ++k) {
    const int dx = k / 9, dy = (k / 3) % 3, dz = k % 3;
    const int ox = x - dx, oy = y - dy, oz = z - dz;
    if (ox >= 0 && ox < SP0 && oy >= 0 && oy < SP1 && oz >= 0 && oz < SP2)
      atomicAdd(&cnt[ox * (SP1 * SP2) + oy * SP2 + oz], 1u);
  }
}

// ---------------- sorted compaction ----------------

__global__ void scan_count(const unsigned* __restrict__ cnt,
                           unsigned* __restrict__ bsum) {
  __shared__ unsigned s[SCAN_THREADS];
  const int t = threadIdx.x;
  long long base = (long long)blockIdx.x * SCAN_CELLS + (long long)t * 16;
  unsigned c = 0;
  for (int j = 0; j < 16; ++j) {
    long long g = base + j;
    if (g < GRIDC) c += (cnt[g] > 0u);
  }
  s[t] = c; __syncthreads();
  for (int off = 128; off > 0; off >>= 1) {
    if (t < off) s[t] += s[t + off];
    __syncthreads();
  }
  if (t == 0) bsum[blockIdx.x] = s[0];
}

__global__ void scan_offsets(const unsigned* __restrict__ bsum,
                             unsigned* __restrict__ boff, int nb) {
  if (blockIdx.x == 0 && threadIdx.x == 0) {
    unsigned acc = 0;
    for (int i = 0; i < nb; ++i) { boff[i] = acc; acc += bsum[i]; }
    boff[nb] = acc;
  }
}

// Writes sorted uniq, converts cnt grid into cell->compacted-row map in
// place, and seeds every occupied output row with the bias vector.
__global__ void compact_map(unsigned* __restrict__ cnt,
                            const unsigned* __restrict__ boff,
                            const float* __restrict__ bias,
                            float* __restrict__ out_feats,
                            int* __restrict__ out_uniq) {
  __shared__ unsigned s[SCAN_THREADS];
  const int t = threadIdx.x;
  long long base = (long long)blockIdx.x * SCAN_CELLS + (long long)t * 16;

  unsigned c = 0;
  for (int j = 0; j < 16; ++j) {
    long long g = base + j;
    if (g < GRIDC) c += (cnt[g] > 0u);
  }
  s[t] = c; __syncthreads();
  // inclusive Hillis-Steele scan over 256 thread counts
  for (int off = 1; off < SCAN_THREADS; off <<= 1) {
    unsigned v = (t >= off) ? s[t - off] : 0u;
    __syncthreads();
    s[t] += v;
    __syncthreads();
  }
  unsigned pos = boff[blockIdx.x] + s[t] - c;   // exclusive prefix

  const float4 b0 = ((const float4*)bias)[0];
  const float4 b1 = ((const float4*)bias)[1];
  const float4 b2 = ((const float4*)bias)[2];
  const float4 b3 = ((const float4*)bias)[3];

  for (int j = 0; j < 16; ++j) {
    long long g = base + j;
    if (g < GRIDC) {
      if (cnt[g] > 0u) {
        cnt[g] = pos;                               // becomes the map
        out_uniq[pos] = (int)g;
        float4* dst = (float4*)(out_feats + (size_t)pos * COUT);
        dst[0] = b0; dst[1] = b1; dst[2] = b2; dst[3] = b3;
        ++pos;
      }
    }
  }
}

// ---------------- WMMA GEMM + atomic scatter into final rows ----------------

__launch_bounds__(256, 2)
__global__ void conv_wmma_scatter(const float* __restrict__ feats,
                                  const int*   __restrict__ coords,
                                  const float* __restrict__ W,
                                  const unsigned* __restrict__ map,
                                  float* __restrict__ out_feats) {
  __shared__ float w_s[KVOL * CIN * COUT];    // 27*64*16 f32 = 108 KB
  __shared__ float f_s[WPB][16][68];          // padded stride 68 -> no bank conflicts

  const int t    = threadIdx.x;
  const int lane = t & 31;
  const int wid  = t >> 5;
  const int tile = blockIdx.x * WPB + wid;

  // async-DMA all 27 weight matrices into LDS (block-wide, 6912 x b128)
  for (int i = t; i < (KVOL * CIN * COUT) / 4; i += 256)
    async_ld_b128(lds_off(&w_s[i * 4]), W + (size_t)i * 4);

  // async-DMA this wave's 16x64 feature tile into padded LDS
  if (tile < TILES) {
    const float* src = feats + (size_t)tile * 16 * CIN;
    for (int j = 0; j < 8; ++j) {
      int i = j * 128 + lane * 4;             // element index in [0,1024)
      int m = i >> 6, c = i & 63;
      async_ld_b128(lds_off(&f_s[wid][m][c]), src + i);
    }
  }
  wait_async0();                              // s_wait_asynccnt 0
  __syncthreads();
  if (tile >= TILES) return;

  // lanes 0..15 own one point each: coords -> cell -> compacted row
  int pos = -1;
  int ptx = 0, pty = 0, ptz = 0;
  if (lane < 16) {
    const int* cp = coords + (size_t)(tile * 16 + lane) * 4;
    ptx = cp[1]; pty = cp[2]; ptz = cp[3];
  }

  const int n  = lane & 15;            // column (output channel)
  const int kh = (lane >> 4) << 1;     // K sub-offset: 0 (lanes 0-15) or 2 (16-31)

  for (int k = 0; k < KVOL; ++k) {
    const int dx = k / 9, dy = (k / 3) % 3, dz = k % 3;

    pos = -1;
    if (lane < 16) {
      const int ox = ptx - dx, oy = pty - dy, oz = ptz - dz;
      const bool valid = (ox >= 0) & (ox < SP0) & (oy >= 0) & (oy < SP1) &
                         (oz >= 0) & (oz < SP2);
      if (valid)
        pos = (int)map[ox * (SP1 * SP2) + oy * SP2 + oz];
    }

    // D(16x16) = A(16x64) * B(64x16) via 16 chained f32 16x16x4 WMMAs
    v8f acc = {};
    const float* wk = &w_s[k * CIN * COUT];
#pragma unroll
    for (int kk = 0; kk < 16; ++kk) {
      const int k0 = kk * 4;
      v2f a, b;
      a.x = f_s[wid][n][k0 + kh];
      a.y = f_s[wid][n][k0 + kh + 1];
      b.x = wk[(k0 + kh)     * COUT + n];
      b.y = wk[(k0 + kh + 1) * COUT + n];
      acc = __builtin_amdgcn_wmma_f32_16x16x4_f32(
          /*neg_a=*/false, a, /*neg_b=*/false, b,
          /*c_mod=*/(short)0, acc, /*reuse_a=*/false, /*reuse_b=*/false);
    }

    // scatter: VGPR r holds row r (lanes 0-15) and row r+8 (lanes 16-31)
#pragma unroll
    for (int r = 0; r < 8; ++r) {
      const int row  = (lane < 16) ? r : (r + 8);
      const int posr = __shfl(pos, row, 32);
      if (posr >= 0)
        atomicAdd(&out_feats[(size_t)posr * COUT + n], acc[r]);
    }
  }
}

// ---------------- launcher ----------------

extern "C" void kernel_launch(void* const* d_in, const int* in_sizes, int n_in,
                              void* d_out, int out_size, void* d_ws, size_t ws_size,
                              hipStream_t stream) {
  const float* feats  = (const float*)d_in[0];
  const int*   coords = (const int*)  d_in[1];
  const float* W      = (const float*)d_in[2];
  const float* bias   = (const float*)d_in[3];

  // workspace layout: [cnt/map: GRIDC u32][bsum: NB][boff: NB+1]  (~152 MB)
  unsigned* cnt  = (unsigned*)d_ws;
  unsigned* bsum = cnt + GRIDC;
  unsigned* boff = bsum + NB;

  float* out_feats = (float*)d_out;
  int*   out_uniq  = ((int*)d_out) + NKTOT * 16;

  zero_cnt_kernel<<<2048, 256, 0, stream>>>((uint4*)cnt, GRIDC / 4);
  fill_out_kernel<<<4096, 256, 0, stream>>>((int*)d_out, NKTOT * 16, NKTOT * 17,
                                            SENTV);
  count_pass<<<(NPTS + 255) / 256, 256, 0, stream>>>(coords, cnt);
  scan_count<<<NB, SCAN_THREADS, 0, stream>>>(cnt, bsum);
  scan_offsets<<<1, 1, 0, stream>>>(bsum, boff, NB);
  compact_map<<<NB, SCAN_THREADS, 0, stream>>>(cnt, boff, bias, out_feats,
                                               out_uniq);
  conv_wmma_scatter<<<(TILES + WPB - 1) / WPB, 256, 0, stream>>>(
      feats, coords, W, cnt /* now the map */, out_feats);
}